// ContextQueryAttention_26560077758672
// MI455X (gfx1250) — compile-verified
//
#include <hip/hip_runtime.h>
#include <math.h>
#include <stdint.h>

// Problem dims (fixed by reference)
#define NB 64
#define CD 1024
#define QD 128
#define FD 512
#define CB 64            // context rows handled per block in k1
#define KC 32            // K-chunk staged through LDS in GEMM1
#define AST 36           // padded K-stride (floats) for staged A/B tiles:
                         //  - 36*4=144 B rows keep 16B alignment for b128 DMA
                         //  - (36*r + k) % 64 banks are distinct for r=0..15 -> conflict-free b64 frags
#define LDS_STRIDE 132   // sim tile stride: 128 + 4 floats pad

typedef float v2f __attribute__((ext_vector_type(2)));
typedef float v8f __attribute__((ext_vector_type(8)));
typedef int   v4i __attribute__((vector_size(4 * sizeof(int))));

using gv4i = __attribute__((address_space(1))) v4i;  // global (prints as __device__)
using lv4i = __attribute__((address_space(3))) v4i;  // LDS

// ---- CDNA5 async Global->LDS copy (ASYNCcnt path, CDNA5 ISA) ----
__device__ __forceinline__ void async_copy16(const float* __restrict__ g,
                                             float* l) {
#if __has_builtin(__builtin_amdgcn_global_load_async_to_lds_b128)
  __builtin_amdgcn_global_load_async_to_lds_b128(
      (gv4i*)(uintptr_t)g, (lv4i*)(uintptr_t)l, 0, 0);
#else
  const uint32_t lds_off = (uint32_t)(uintptr_t)l;  // low 32 bits = LDS byte offset
  asm volatile("global_load_async_to_lds_b128 %0, %1, off"
               :: "v"(lds_off), "v"(g)
               : "memory");
#endif
}

__device__ __forceinline__ void wait_async0() {
#if __has_builtin(__builtin_amdgcn_s_wait_asynccnt)
  __builtin_amdgcn_s_wait_asynccnt(0);
#else
  asm volatile("s_wait_asynccnt 0x0" ::: "memory");
#endif
}

// -------------------------------------------------------------------------
// k0: s_ctx[n,c] = <x_context[n,c,:], w_context>, s_qry[n,q] = <x_query[n,q,:], w_context>
// wave-per-row, float4 loads, wave32 butterfly reduction.
// -------------------------------------------------------------------------
__global__ __launch_bounds__(256) void rowdot_kernel(
    const float* __restrict__ xc, const float* __restrict__ xq,
    const float* __restrict__ wc, float* __restrict__ sctx,
    float* __restrict__ sqry) {
  const int row  = blockIdx.x * 8 + (threadIdx.x >> 5);
  const int lane = threadIdx.x & 31;
  const float* src;
  float* dst;
  if (row < NB * CD) {
    src = xc + (size_t)row * FD;
    dst = sctx + row;
  } else {
    const int r = row - NB * CD;
    src = xq + (size_t)r * FD;
    dst = sqry + r;
  }
  float acc = 0.f;
  for (int f = lane * 4; f < FD; f += 32 * 4) {
    float4 a = *(const float4*)(src + f);
    float4 w = *(const float4*)(wc + f);
    acc += a.x * w.x + a.y * w.y + a.z * w.z + a.w * w.w;
  }
#pragma unroll
  for (int m = 16; m >= 1; m >>= 1) acc += __shfl_xor(acc, m, 32);
  if (lane == 0) *dst = acc;
}

// -------------------------------------------------------------------------
// k1: per (n, 64-row C strip):
//   GEMM1 (WMMA f32 16x16x4, LDS-staged via async DMA):
//       sim[64x128] = (x_ctx * w_cq) @ x_query^T + s_ctx + s_qry
//   row softmax over q (rowmax -> workspace)
//   GEMM2 (WMMA f32 16x16x4): x_c2q[64x512] = P @ x_query -> d_out slice [F..2F)
// -------------------------------------------------------------------------
__global__ __launch_bounds__(256) void sim_softmax_c2q_kernel(
    const float* __restrict__ xc, const float* __restrict__ xq,
    const float* __restrict__ wcq, const float* __restrict__ sctx,
    const float* __restrict__ sqry, float* __restrict__ rowmax,
    float* __restrict__ out) {
  __shared__ float wLds[FD];               //  2048 B
  __shared__ float aLds[CB * AST];         //  9216 B  x_ctx chunk [64 x 32]
  __shared__ float bLds[QD * AST];         // 18432 B  x_qry chunk [128 x 32]
  __shared__ float simld[CB * LDS_STRIDE]; // 33792 B  sim / P tile

  const int blk   = blockIdx.x;              // 0 .. N*(C/CB)-1
  const int n     = blk / (CD / CB);
  const int cb    = (blk % (CD / CB)) * CB;  // base context row
  const int tid   = threadIdx.x;
  const int wave  = tid >> 5;
  const int lane  = tid & 31;
  const int lhalf = lane >> 4;  // selects K-pair (A/B frags) / M+8 (C frag)
  const int lmod  = lane & 15;

  const float* xcn = xc + (size_t)n * CD * FD;
  const float* xqn = xq + (size_t)n * QD * FD;

  // stage w_cq once (visible after the first chunk barrier)
  wLds[tid]       = wcq[tid];
  wLds[tid + 256] = wcq[tid + 256];

  // ---------------- GEMM1: LDS-staged WMMA over F ----------------
  {
    const int mt = wave & 3;   // m-tile 0..3
    const int ng = wave >> 2;  // 0|1 -> n-tiles ng*4 .. ng*4+3
    const int ar = mt * 16 + lmod;  // A-frag local row

    v8f acc[4];
#pragma unroll
    for (int t = 0; t < 4; ++t) acc[t] = (v8f){0, 0, 0, 0, 0, 0, 0, 0};

    for (int c = 0; c < FD / KC; ++c) {
      const int k0 = c * KC;
      // ---- async DMA stage: A chunk (2 x b128/thread), B chunk (4 x b128/thread)
      {
        const int e0 = tid, e1 = tid + 256;  // 512 float4s of A
        async_copy16(xcn + (size_t)(cb + (e0 >> 3)) * FD + k0 + (e0 & 7) * 4,
                     &aLds[(e0 >> 3) * AST + (e0 & 7) * 4]);
        async_copy16(xcn + (size_t)(cb + (e1 >> 3)) * FD + k0 + (e1 & 7) * 4,
                     &aLds[(e1 >> 3) * AST + (e1 & 7) * 4]);
      }
#pragma unroll
      for (int i = 0; i < 4; ++i) {        // 1024 float4s of B
        const int e = tid + i * 256;
        async_copy16(xqn + (size_t)(e >> 3) * FD + k0 + (e & 7) * 4,
                     &bLds[(e >> 3) * AST + (e & 7) * 4]);
      }
      wait_async0();
      __syncthreads();

      // ---- compute chunk from LDS (conflict-free b64 frag reads)
#pragma unroll
      for (int kk = 0; kk < KC; kk += 4) {
        const int ka = kk + 2 * lhalf;
        v2f a = *(const v2f*)(&aLds[ar * AST + ka]);
        v2f w = *(const v2f*)(&wLds[k0 + ka]);
        a.x *= w.x;
        a.y *= w.y;  // A' = x_ctx * w_cq
#pragma unroll
        for (int t = 0; t < 4; ++t) {
          v2f b = *(const v2f*)(&bLds[((ng * 4 + t) * 16 + lmod) * AST + ka]);
          acc[t] = __builtin_amdgcn_wmma_f32_16x16x4_f32(
              false, a, false, b, (short)0, acc[t], false, false);
        }
      }
      __syncthreads();  // chunk consumed; safe to overwrite stage buffers
    }

    // bias add + stage sim to LDS (C/D layout: vgpr r -> M=r+8*lhalf, N=lmod)
    float sq[4];
#pragma unroll
    for (int t = 0; t < 4; ++t)
      sq[t] = sqry[(size_t)n * QD + (ng * 4 + t) * 16 + lmod];
#pragma unroll
    for (int r = 0; r < 8; ++r) {
      const int mrow = mt * 16 + r + 8 * lhalf;  // local row 0..63
      const float sc = sctx[(size_t)n * CD + cb + mrow];
#pragma unroll
      for (int t = 0; t < 4; ++t) {
        const int col = (ng * 4 + t) * 16 + lmod;  // 0..127
        simld[mrow * LDS_STRIDE + col] = acc[t][r] + sc + sq[t];
      }
    }
  }
  __syncthreads();

  // ---------------- softmax over q per row (waves 0,1 fully active) -------
  if (tid < CB) {
    float* rowp = &simld[tid * LDS_STRIDE];
    float m = -INFINITY;
    for (int j = 0; j < QD; ++j) m = fmaxf(m, rowp[j]);
    rowmax[(size_t)n * CD + cb + tid] = m;  // feeds q2c path
    float s = 0.f;
    for (int j = 0; j < QD; ++j) {
      float e = __expf(rowp[j] - m);
      rowp[j] = e;
      s += e;
    }
    const float inv = 1.f / s;
    for (int j = 0; j < QD; ++j) rowp[j] *= inv;
  }
  __syncthreads();

  // ---------------- GEMM2: x_c2q = P @ x_query, P from LDS ----------------
  {
    const int mt   = wave & 3;
    const int ntb  = (wave >> 2) * 16;  // 16 f-tiles per wave
    const int mrow = mt * 16 + lmod;    // A-frag row
    for (int t = 0; t < 16; ++t) {
      const int f0 = (ntb + t) * 16;
      v8f acc = (v8f){0, 0, 0, 0, 0, 0, 0, 0};
      for (int k = 0; k < QD; k += 4) {
        const int ka = k + 2 * lhalf;
        v2f a = *(const v2f*)(&simld[mrow * LDS_STRIDE + ka]);  // ds_load_b64
        v2f b;
        b.x = xqn[(size_t)ka * FD + f0 + lmod];
        b.y = xqn[(size_t)(ka + 1) * FD + f0 + lmod];
        acc = __builtin_amdgcn_wmma_f32_16x16x4_f32(
            false, a, false, b, (short)0, acc, false, false);
      }
#pragma unroll
      for (int r = 0; r < 8; ++r) {
        const int lr = mt * 16 + r + 8 * lhalf;
        const size_t o =
            ((size_t)n * CD + cb + lr) * (size_t)(4 * FD) + FD + f0 + lmod;
        out[o] = acc[r];  // x_c2q lands in output slice [F..2F)
      }
    }
  }
}

// -------------------------------------------------------------------------
// k2: q2c_w = softmax_c(rowmax[n,:]); x_q2c[n,f] = sum_c q2c_w[c]*x_ctx[n,c,f]
// -------------------------------------------------------------------------
__global__ __launch_bounds__(256) void q2c_kernel(
    const float* __restrict__ xc, const float* __restrict__ rowmax,
    float* __restrict__ xq2c) {
  __shared__ float red[256];
  __shared__ float wl[CD];
  const int n = blockIdx.x;
  const int t = threadIdx.x;
  const float* rm = rowmax + (size_t)n * CD;

  float m = -INFINITY;
  for (int c = t; c < CD; c += 256) m = fmaxf(m, rm[c]);
  red[t] = m;
  __syncthreads();
  for (int s = 128; s > 0; s >>= 1) {
    if (t < s) red[t] = fmaxf(red[t], red[t + s]);
    __syncthreads();
  }
  m = red[0];
  __syncthreads();

  float s = 0.f;
  for (int c = t; c < CD; c += 256) {
    float e = __expf(rm[c] - m);
    wl[c] = e;
    s += e;
  }
  red[t] = s;
  __syncthreads();
  for (int st = 128; st > 0; st >>= 1) {
    if (t < st) red[t] += red[t + st];
    __syncthreads();
  }
  const float inv = 1.f / red[0];
  __syncthreads();

  const float* x = xc + (size_t)n * CD * FD;
  float a0 = 0.f, a1 = 0.f;
  for (int c = 0; c < CD; ++c) {
    const float w = wl[c] * inv;
    a0 += w * x[(size_t)c * FD + t];  // coalesced across threads
    a1 += w * x[(size_t)c * FD + t + 256];
  }
  xq2c[(size_t)n * FD + t] = a0;
  xq2c[(size_t)n * FD + t + 256] = a1;
}

// -------------------------------------------------------------------------
// k3: epilogue. out[n,c,:] = [x_ctx | x_c2q | x_ctx*x_c2q | x_ctx*x_q2c]
// x_c2q already resides in out slice [F..2F). One block per (n,c).
// -------------------------------------------------------------------------
__global__ __launch_bounds__(256) void epilogue_kernel(
    const float* __restrict__ xc, const float* __restrict__ xq2c,
    float* __restrict__ out) {
  const int nc = blockIdx.x;  // n*C + c
  const int n  = nc / CD;
  const int f  = threadIdx.x * 2;

  const float2 cv  = *(const float2*)(xc + (size_t)nc * FD + f);
  float2* orow     = (float2*)(out + (size_t)nc * (size_t)(4 * FD));
  const float2 c2q = orow[(FD + f) >> 1];
  const float2 q2c = *(const float2*)(xq2c + (size_t)n * FD + f);

  orow[f >> 1] = cv;                                                  // slice 0
  orow[(2 * FD + f) >> 1] = make_float2(cv.x * c2q.x, cv.y * c2q.y);  // slice 2
  orow[(3 * FD + f) >> 1] = make_float2(cv.x * q2c.x, cv.y * q2c.y);  // slice 3
}

// -------------------------------------------------------------------------
extern "C" void kernel_launch(void* const* d_in, const int* in_sizes, int n_in,
                              void* d_out, int out_size, void* d_ws,
                              size_t ws_size, hipStream_t stream) {
  const float* xc  = (const float*)d_in[0];  // [N,C,F]
  const float* xq  = (const float*)d_in[1];  // [N,Q,F]
  const float* wc  = (const float*)d_in[2];  // [F]
  // d_in[3] = w_query: declared-but-unused in the reference
  const float* wcq = (const float*)d_in[4];  // [F]
  float* out = (float*)d_out;                // [N,C,4F]

  // workspace layout (floats): sctx | sqry | rowmax | xq2c  (~688 KB total)
  float* ws     = (float*)d_ws;
  float* sctx   = ws;                      // N*C = 65536
  float* sqry   = ws + NB * CD;            // N*Q = 8192
  float* rmax   = ws + NB * CD + NB * QD;  // N*C = 65536
  float* xq2c_w = rmax + NB * CD;          // N*F = 32768

  rowdot_kernel<<<(NB * CD + NB * QD) / 8, 256, 0, stream>>>(xc, xq, wc, sctx,
                                                             sqry);
  sim_softmax_c2q_kernel<<<NB * (CD / CB), 256, 0, stream>>>(
      xc, xq, wcq, sctx, sqry, rmax, out);
  q2c_kernel<<<NB, 256, 0, stream>>>(xc, rmax, xq2c_w);
  epilogue_kernel<<<NB * CD, 256, 0, stream>>>(xc, xq2c_w, out);
}